// StateGNNEncoderConvEdgeAttrCompact_62955630624977
// MI455X (gfx1250) — compile-verified
//
#include <hip/hip_runtime.h>
#include <hip/hip_bf16.h>

typedef __attribute__((ext_vector_type(2))) float v2f;
typedef __attribute__((ext_vector_type(8))) float v8f;

// ---------------------------------------------------------------------------
// Elementwise / scatter helpers
// ---------------------------------------------------------------------------

__global__ void deg_kernel(const long long* __restrict__ col, int E,
                           float* __restrict__ deg) {
  int e = blockIdx.x * blockDim.x + threadIdx.x;
  if (e < E) atomicAdd(deg + col[e], 1.0f);
}

__global__ void rsqrt_kernel(float* __restrict__ d, int N) {
  int t = blockIdx.x * blockDim.x + threadIdx.x;
  if (t < N) {
    float v = d[t];
    d[t] = (v > 0.0f) ? rsqrtf(v) : 0.0f;
  }
}

// norm-weighted hop for small feature dim C (TAGConv1 on 5-dim game features)
template <int C>
__global__ void tag_hop_small_kernel(const long long* __restrict__ row,
                                     const long long* __restrict__ col,
                                     const float* __restrict__ dinv,
                                     const float* __restrict__ xin,
                                     float* __restrict__ xout, int E) {
  int e = blockIdx.x * blockDim.x + threadIdx.x;
  if (e >= E) return;
  long long r = row[e], c = col[e];
  float nm = dinv[r] * dinv[c];
  const float* sp = xin + r * C;
  float* dp = xout + c * C;
#pragma unroll
  for (int k = 0; k < C; ++k) atomicAdd(dp + k, nm * sp[k]);
}

// norm-weighted hop, 64 channels: thread per (edge, quad-of-4-channels)
__global__ void tag_hop64_kernel(const long long* __restrict__ row,
                                 const long long* __restrict__ col,
                                 const float* __restrict__ dinv,
                                 const float* __restrict__ xin,
                                 float* __restrict__ xout, int E) {
  int t = blockIdx.x * blockDim.x + threadIdx.x;
  if (t >= E * 16) return;
  int e = t >> 4;
  int q = (t & 15) << 2;
  long long r = row[e], c = col[e];
  float nm = dinv[r] * dinv[c];
  const float4 v = *(const float4*)(xin + r * 64 + q);
  float* dp = xout + c * 64 + q;
  atomicAdd(dp + 0, nm * v.x);
  atomicAdd(dp + 1, nm * v.y);
  atomicAdd(dp + 2, nm * v.z);
  atomicAdd(dp + 3, nm * v.w);
}

// GraphConv scatter: dst[col] += ew * src[row]
__global__ void scatter_ew_kernel(const long long* __restrict__ row,
                                  const long long* __restrict__ col,
                                  const float* __restrict__ ew,
                                  const float* __restrict__ src,
                                  float* __restrict__ dst, int E) {
  int t = blockIdx.x * blockDim.x + threadIdx.x;
  if (t >= E * 16) return;
  int e = t >> 4;
  int q = (t & 15) << 2;
  long long r = row[e], c = col[e];
  float w = ew[e];
  const float4 v = *(const float4*)(src + r * 64 + q);
  float* dp = dst + c * 64 + q;
  atomicAdd(dp + 0, w * v.x);
  atomicAdd(dp + 1, w * v.y);
  atomicAdd(dp + 2, w * v.z);
  atomicAdd(dp + 3, w * v.w);
}

// SAGE scatter: s[col] += src[row]; cnt[col] += 1
__global__ void scatter_sum_kernel(const long long* __restrict__ row,
                                   const long long* __restrict__ col,
                                   const float* __restrict__ src,
                                   float* __restrict__ s,
                                   float* __restrict__ cnt, int E) {
  int t = blockIdx.x * blockDim.x + threadIdx.x;
  if (t >= E * 16) return;
  int e = t >> 4;
  int q = (t & 15) << 2;
  long long r = row[e], c = col[e];
  const float4 v = *(const float4*)(src + r * 64 + q);
  float* dp = s + c * 64 + q;
  atomicAdd(dp + 0, v.x);
  atomicAdd(dp + 1, v.y);
  atomicAdd(dp + 2, v.z);
  atomicAdd(dp + 3, v.w);
  if (q == 0) atomicAdd(cnt + c, 1.0f);
}

__global__ void mean_div_kernel(float* __restrict__ s,
                                const float* __restrict__ cnt, int N) {
  int t = blockIdx.x * blockDim.x + threadIdx.x;
  if (t >= N * 64) return;
  int n = t >> 6;
  s[t] = s[t] / fmaxf(cnt[n], 1.0f);
}

// TAGConv1 combine: out = relu(x@W0 + h1@W1 + h2@W2 + b), K=5, OUT=64
__global__ void tag1_combine_kernel(const float* __restrict__ x,
                                    const float* __restrict__ h1,
                                    const float* __restrict__ h2,
                                    const float* __restrict__ W,
                                    const float* __restrict__ b,
                                    float* __restrict__ out, int N) {
  int t = blockIdx.x * blockDim.x + threadIdx.x;
  if (t >= N * 64) return;
  int n = t >> 6, c = t & 63;
  const float* W0 = W;
  const float* W1 = W + 5 * 64;
  const float* W2 = W + 10 * 64;
  const float* xr = x + n * 5;
  const float* h1r = h1 + n * 5;
  const float* h2r = h2 + n * 5;
  float r = b[c];
#pragma unroll
  for (int k = 0; k < 5; ++k)
    r += xr[k] * W0[k * 64 + c] + h1r[k] * W1[k * 64 + c] +
         h2r[k] * W2[k * 64 + c];
  out[t] = fmaxf(r, 0.0f);
}

// final: out[N x 8] = x[N x 64] @ W[64 x 8] + b
__global__ void final_linear_kernel(const float* __restrict__ x,
                                    const float* __restrict__ W,
                                    const float* __restrict__ b,
                                    float* __restrict__ out, int N) {
  int t = blockIdx.x * blockDim.x + threadIdx.x;
  if (t >= N * 8) return;
  int n = t >> 3, o = t & 7;
  const float* xr = x + n * 64;
  float r = b[o];
#pragma unroll
  for (int k = 0; k < 64; ++k) r += xr[k] * W[k * 8 + o];
  out[t] = r;
}

// ---------------------------------------------------------------------------
// WMMA GEMM: Y[nrows x 64] = post( X@W (+ X2@W2) (+ Xs@Ws, K=CS) (+ addend)
//                                  (+ bias) ), post = optional relu.
// fp32-exact via V_WMMA_F32_16X16X4_F32. One wave -> 16x64 output tile.
// Block = 256 threads (8 waves) -> 128 rows. nrows must be multiple of 128.
// ---------------------------------------------------------------------------
template <bool HAS_X2, int CS, bool HAS_ADD, bool HAS_BIAS, bool RELU>
__global__ __launch_bounds__(256) void gemm64_kernel(
    const float* __restrict__ X, const float* __restrict__ W,
    const float* __restrict__ X2, const float* __restrict__ W2,
    const float* __restrict__ Xs, const float* __restrict__ Ws,
    const float* __restrict__ addend, const float* __restrict__ bias,
    float* __restrict__ Y, int nrows) {
  __shared__ float sW[64 * 64];
  __shared__ float sW2[HAS_X2 ? 64 * 64 : 1];
  __shared__ float sWs[(CS > 0) ? CS * 64 : 1];

  int tid = threadIdx.x;
  for (int i = tid; i < 64 * 64; i += 256) sW[i] = W[i];
  if constexpr (HAS_X2)
    for (int i = tid; i < 64 * 64; i += 256) sW2[i] = W2[i];
  if constexpr (CS > 0)
    for (int i = tid; i < CS * 64; i += 256) sWs[i] = Ws[i];
  __syncthreads();

  int wave = tid >> 5;
  int lane = tid & 31;
  int laneN = lane & 15;  // A: row-in-tile; B/D: column-in-tile
  int hi = lane >> 4;     // selects K pair (A/B) / row half (C,D)
  int m0 = (blockIdx.x * 8 + wave) * 16;

  v8f c[4] = {};

  // --- main dense GEMM: X @ W ---
  {
    const float* Xrow = X + (size_t)(m0 + laneN) * 64;
#pragma unroll
    for (int k0 = 0; k0 < 64; k0 += 4) {
      v2f a;
      a.x = Xrow[k0 + 2 * hi];
      a.y = Xrow[k0 + 2 * hi + 1];
#pragma unroll
      for (int j = 0; j < 4; ++j) {
        v2f bb;
        bb.x = sW[(k0 + 2 * hi) * 64 + j * 16 + laneN];
        bb.y = sW[(k0 + 2 * hi + 1) * 64 + j * 16 + laneN];
        c[j] = __builtin_amdgcn_wmma_f32_16x16x4_f32(
            false, a, false, bb, (short)0, c[j], false, false);
      }
    }
  }

  // --- optional second dense GEMM: X2 @ W2 ---
  if constexpr (HAS_X2) {
    const float* Xrow = X2 + (size_t)(m0 + laneN) * 64;
#pragma unroll
    for (int k0 = 0; k0 < 64; k0 += 4) {
      v2f a;
      a.x = Xrow[k0 + 2 * hi];
      a.y = Xrow[k0 + 2 * hi + 1];
#pragma unroll
      for (int j = 0; j < 4; ++j) {
        v2f bb;
        bb.x = sW2[(k0 + 2 * hi) * 64 + j * 16 + laneN];
        bb.y = sW2[(k0 + 2 * hi + 1) * 64 + j * 16 + laneN];
        c[j] = __builtin_amdgcn_wmma_f32_16x16x4_f32(
            false, a, false, bb, (short)0, c[j], false, false);
      }
    }
  }

  // --- epilogue ---
#pragma unroll
  for (int v = 0; v < 8; ++v) {
    int m = m0 + v + 8 * hi;
#pragma unroll
    for (int j = 0; j < 4; ++j) {
      int n = j * 16 + laneN;
      float r = c[j][v];
      if constexpr (HAS_ADD) r += addend[(size_t)m * 64 + n];
      if constexpr (CS > 0) {
        const float* xs = Xs + (size_t)m * CS;
#pragma unroll
        for (int k = 0; k < CS; ++k) r += xs[k] * sWs[k * 64 + n];
      }
      if constexpr (HAS_BIAS) r += bias[n];
      if constexpr (RELU) r = fmaxf(r, 0.0f);
      Y[(size_t)m * 64 + n] = r;
    }
  }
}

// ---------------------------------------------------------------------------

static inline int cdiv(int a, int b) { return (a + b - 1) / b; }

extern "C" void kernel_launch(void* const* d_in, const int* in_sizes, int n_in,
                              void* d_out, int out_size, void* d_ws,
                              size_t ws_size, hipStream_t stream) {
  const float* x_game = (const float*)d_in[0];        // [N,5]
  const float* x_state = (const float*)d_in[1];       // [N,6]
  const long long* e_gg = (const long long*)d_in[2];  // [2,EGG] int64
  const long long* e_h = (const long long*)d_in[3];   // [2,EH]
  const long long* e_in = (const long long*)d_in[4];  // [2,EIN]
  const long long* e_ss = (const long long*)d_in[5];  // [2,ESS]
  const float* ew_h = (const float*)d_in[6];          // [EH]
  const float* tag1_w = (const float*)d_in[7];        // [3,5,64]
  const float* tag1_b = (const float*)d_in[8];        // [64]
  const float* tag2_w = (const float*)d_in[9];        // [4,64,64]
  const float* tag2_b = (const float*)d_in[10];       // [64]
  const float* gc_w_rel = (const float*)d_in[11];     // [64,64]
  const float* gc_b_rel = (const float*)d_in[12];     // [64]
  const float* gc_w_root = (const float*)d_in[13];    // [6,64]
  const float* sage_w_l = (const float*)d_in[14];     // [64,64]
  const float* sage_b_l = (const float*)d_in[15];     // [64]
  const float* sage_w_r = (const float*)d_in[16];     // [64,64]
  const float* lin_w = (const float*)d_in[17];        // [64,8]
  const float* lin_b = (const float*)d_in[18];        // [8]

  const int N = in_sizes[0] / 5;
  const int EGG = in_sizes[2] / 2;
  const int EH = in_sizes[3] / 2;
  const int EIN = in_sizes[4] / 2;
  const int ESS = in_sizes[5] / 2;

  // workspace layout (floats)
  float* ws = (float*)d_ws;
  const size_t NB = (size_t)N * 64;
  float* game_x = ws;            // [N,64]  (later: TAG2 hop buf xk1)
  float* bufP = ws + NB;         // h1 [N,5] / state1 / TAG2 xk2
  float* bufQ = ws + 2 * NB;     // h2 [N,5] / state2 / final state
  float* acc = ws + 3 * NB;      // TAG2 accumulator
  float* agg = ws + 4 * NB;      // scatter target / mean / TAG2 xk3
  float* dinv = ws + 5 * NB;     // [N]
  float* cnt = dinv + N;         // [N]

  const int T = 256;
  const int gemmBlocks = N / 128;  // N = 262144 -> 2048, exact

  // ================= TAGConv1 (game graph, K=2, 5 -> 64) =================
  hipMemsetAsync(dinv, 0, (size_t)N * 4, stream);
  deg_kernel<<<cdiv(EGG, T), T, 0, stream>>>(e_gg + EGG, EGG, dinv);
  rsqrt_kernel<<<cdiv(N, T), T, 0, stream>>>(dinv, N);
  hipMemsetAsync(bufP, 0, (size_t)N * 5 * 4, stream);
  hipMemsetAsync(bufQ, 0, (size_t)N * 5 * 4, stream);
  tag_hop_small_kernel<5><<<cdiv(EGG, T), T, 0, stream>>>(
      e_gg, e_gg + EGG, dinv, x_game, bufP, EGG);
  tag_hop_small_kernel<5><<<cdiv(EGG, T), T, 0, stream>>>(
      e_gg, e_gg + EGG, dinv, bufP, bufQ, EGG);
  tag1_combine_kernel<<<cdiv(N * 64, T), T, 0, stream>>>(
      x_game, bufP, bufQ, tag1_w, tag1_b, game_x, N);

  // ================= GraphConv (hist edges, game -> state) ================
  hipMemsetAsync(agg, 0, NB * 4, stream);
  scatter_ew_kernel<<<cdiv(EH * 16, T), T, 0, stream>>>(
      e_h, e_h + EH, ew_h, game_x, agg, EH);
  // state1 = relu(agg @ w_rel + b_rel + x_state @ w_root)
  gemm64_kernel<false, 6, false, true, true><<<gemmBlocks, T, 0, stream>>>(
      agg, gc_w_rel, nullptr, nullptr, x_state, gc_w_root, nullptr, gc_b_rel,
      bufP, N);

  // ================= SAGEConv (in edges, game -> state) ===================
  hipMemsetAsync(agg, 0, NB * 4, stream);
  hipMemsetAsync(cnt, 0, (size_t)N * 4, stream);
  scatter_sum_kernel<<<cdiv(EIN * 16, T), T, 0, stream>>>(
      e_in, e_in + EIN, game_x, agg, cnt, EIN);
  mean_div_kernel<<<cdiv(N * 64, T), T, 0, stream>>>(agg, cnt, N);
  // state2 = relu(mean @ w_l + b_l + state1 @ w_r)
  gemm64_kernel<true, 0, false, true, true><<<gemmBlocks, T, 0, stream>>>(
      agg, sage_w_l, bufP, sage_w_r, nullptr, nullptr, nullptr, sage_b_l,
      bufQ, N);

  // ================= TAGConv2 (state graph, K=3, 64 -> 64) ================
  hipMemsetAsync(dinv, 0, (size_t)N * 4, stream);
  deg_kernel<<<cdiv(ESS, T), T, 0, stream>>>(e_ss + ESS, ESS, dinv);
  rsqrt_kernel<<<cdiv(N, T), T, 0, stream>>>(dinv, N);

  // acc = state2 @ W0
  gemm64_kernel<false, 0, false, false, false><<<gemmBlocks, T, 0, stream>>>(
      bufQ, tag2_w, nullptr, nullptr, nullptr, nullptr, nullptr, nullptr,
      acc, N);
  // hop 1: xk1 (game_x buf) = Anorm @ state2
  hipMemsetAsync(game_x, 0, NB * 4, stream);
  tag_hop64_kernel<<<cdiv(ESS * 16, T), T, 0, stream>>>(
      e_ss, e_ss + ESS, dinv, bufQ, game_x, ESS);
  gemm64_kernel<false, 0, true, false, false><<<gemmBlocks, T, 0, stream>>>(
      game_x, tag2_w + 1 * 4096, nullptr, nullptr, nullptr, nullptr, acc,
      nullptr, acc, N);
  // hop 2: xk2 (bufP) = Anorm @ xk1
  hipMemsetAsync(bufP, 0, NB * 4, stream);
  tag_hop64_kernel<<<cdiv(ESS * 16, T), T, 0, stream>>>(
      e_ss, e_ss + ESS, dinv, game_x, bufP, ESS);
  gemm64_kernel<false, 0, true, false, false><<<gemmBlocks, T, 0, stream>>>(
      bufP, tag2_w + 2 * 4096, nullptr, nullptr, nullptr, nullptr, acc,
      nullptr, acc, N);
  // hop 3: xk3 (agg) = Anorm @ xk2 ; finalize with bias + relu
  hipMemsetAsync(agg, 0, NB * 4, stream);
  tag_hop64_kernel<<<cdiv(ESS * 16, T), T, 0, stream>>>(
      e_ss, e_ss + ESS, dinv, bufP, agg, ESS);
  gemm64_kernel<false, 0, true, true, true><<<gemmBlocks, T, 0, stream>>>(
      agg, tag2_w + 3 * 4096, nullptr, nullptr, nullptr, nullptr, acc, tag2_b,
      bufQ, N);

  // ================= final linear 64 -> 8 =================================
  final_linear_kernel<<<cdiv(N * 8, T), T, 0, stream>>>(
      bufQ, lin_w, lin_b, (float*)d_out, N);
}